// PacketReceptionProb_15384572854873
// MI455X (gfx1250) — compile-verified
//
#include <hip/hip_runtime.h>
#include <hip/hip_bf16.h>

#define D_ROWS   100000
#define S_COLS   512
#define WINDOW   2000
#define OUT_ROWS (D_ROWS - WINDOW + 1)   // 98001
#define NCHUNK   (D_ROWS / 16)           // 6250
#define NJ0      ((OUT_ROWS + 15) / 16)  // 6126 output groups (j0 = i>>4)
#define INV_W    (1.0f / 2000.0f)

typedef float v2f __attribute__((ext_vector_type(2)));
typedef float v8f __attribute__((ext_vector_type(8)));

// log2(1 - e[d,s]), with rows >= D treated as success==1 -> log2 = 0
__device__ __forceinline__ float lg_succ(const float* __restrict__ e, int d, int s) {
    if (d >= D_ROWS) return 0.0f;
    float x = 1.0f - e[d * S_COLS + s];
    return __log2f(x);
}

// ---------------- Kernel 1: 16-row chunk sums of r ----------------
// r[d,s] = exp2( 125 * sum_{o=0..7} log2(succ[d+o]) + 64 * log2(succ[d+8]) )
__global__ void __launch_bounds__(512)
k_chunk_sums(const float* __restrict__ e, float* __restrict__ cs) {
    const int s  = threadIdx.x;     // 0..511 (coalesced along columns)
    const int j  = blockIdx.x;      // chunk id 0..6249
    const int d0 = j << 4;

    float t[9];
#pragma unroll
    for (int o = 0; o < 9; ++o) t[o] = lg_succ(e, d0 + o, s);
    float s8 = t[0] + t[1] + t[2] + t[3] + t[4] + t[5] + t[6] + t[7];

    float sum = 0.0f;
#pragma unroll
    for (int q = 0; q < 16; ++q) {
        sum += exp2f(fmaf(125.0f, s8, 64.0f * t[8]));
        s8 += t[8] - t[0];
#pragma unroll
        for (int o = 0; o < 8; ++o) t[o] = t[o + 1];
        t[8] = lg_succ(e, d0 + q + 9, s);
    }
    cs[j * S_COLS + s] = sum;
}

// ---------------- Kernel 2: windowed outputs ----------------
// One wave per (j0, 16-column group). Base = sum of 125 chunk sums via
// chained v_wmma_f32_16x16x4_f32 with all-ones A (column-sum broadcast).
__global__ void __launch_bounds__(256)
k_window_out(const float* __restrict__ e, const float* __restrict__ cs,
             float* __restrict__ out) {
    const int lane = threadIdx.x & 31;
    const int wave = threadIdx.x >> 5;
    const int j0   = blockIdx.y * 8 + wave;
    if (j0 >= NJ0) return;                 // wave-uniform: EXEC stays full below

    const int col  = lane & 15;
    const int half = lane >> 4;            // 0: low edge, 1: high (+2000) edge
    const int s    = (blockIdx.x << 4) + col;

    // --- base: sum chunkSum[j0 .. j0+124] on the matrix pipe ---
    v8f c = {};
    v2f a; a.x = 1.0f; a.y = 1.0f;         // A = ones(16x4)
    for (int jj = 0; jj < 124; jj += 4) {  // 31 WMMAs cover 124 chunks
        const int jr = (j0 + jj + 2 * half) * S_COLS + s;
        v2f b;
        b.x = cs[jr];
        b.y = cs[jr + S_COLS];
        c = __builtin_amdgcn_wmma_f32_16x16x4_f32(
                false, a, false, b, (short)0, c, false, false);
    }
    const float base = c[0] + cs[(j0 + 124) * S_COLS + s];

    // --- edge prefix streams: recompute 16 r values (input is L2-resident) ---
    const int d0 = (j0 << 4) + half * WINDOW;
    float t[9];
#pragma unroll
    for (int o = 0; o < 9; ++o) t[o] = lg_succ(e, d0 + o, s);
    float s8 = t[0] + t[1] + t[2] + t[3] + t[4] + t[5] + t[6] + t[7];

    float pref = 0.0f;                      // sum_{q<p} r[d0+q]
#pragma unroll
    for (int p = 0; p < 16; ++p) {
        const float other = __shfl_xor(pref, 16);   // partner half's prefix
        if (half == 0) {
            const int i = (j0 << 4) + p;
            if (i < OUT_ROWS)
                out[i * S_COLS + s] = (base + other - pref) * INV_W;
        }
        pref += exp2f(fmaf(125.0f, s8, 64.0f * t[8]));
        s8 += t[8] - t[0];
#pragma unroll
        for (int o = 0; o < 8; ++o) t[o] = t[o + 1];
        t[8] = lg_succ(e, d0 + p + 9, s);
    }
}

extern "C" void kernel_launch(void* const* d_in, const int* in_sizes, int n_in,
                              void* d_out, int out_size, void* d_ws, size_t ws_size,
                              hipStream_t stream) {
    const float* e = (const float*)d_in[0];
    float* out = (float*)d_out;
    float* cs  = (float*)d_ws;   // 6250 * 512 * 4 B = 12.8 MB scratch

    k_chunk_sums<<<NCHUNK, 512, 0, stream>>>(e, cs);

    dim3 grid(S_COLS / 16, (NJ0 + 7) / 8); // (32, 766)
    k_window_out<<<grid, 256, 0, stream>>>(e, cs, out);
}